// QuaternionRotation_53060025975049
// MI455X (gfx1250) — compile-verified
//
#include <hip/hip_runtime.h>
#include <stdint.h>

#define DGRID 256
#define NTOT  (DGRID * DGRID * DGRID)

typedef uint32_t u32x4 __attribute__((ext_vector_type(4)));
typedef uint32_t u32x8 __attribute__((ext_vector_type(8)));

// ---------------------------------------------------------------------------
// Kernel 1: zero the 64MB output with wide b128 stores (harness poisons d_out).
// ---------------------------------------------------------------------------
__global__ void qrot_zero(float* __restrict__ out) {
    const uint32_t idx = blockIdx.x * blockDim.x + threadIdx.x;  // one float4 each
    float4 z = make_float4(0.f, 0.f, 0.f, 0.f);
    reinterpret_cast<float4*>(out)[idx] = z;
}

// ---------------------------------------------------------------------------
// Kernel 2: TDM-staged quaternion rotate + scatter.
//   - block = 256 threads = 8 waves; block tile = 8192 voxels (32KB LDS)
//   - each wave DMAs its own 4KB (1024-float) slice via tensor_load_to_lds,
//     waits on TENSORcnt (per-wave, so no __syncthreads needed)
//   - each lane processes 8 x float4 from LDS, conflict-free layout
//   - rotation math: fp contract OFF so mul/add rounding matches the unfused
//     f32 reference exactly (result is truncated to int -> ulp-sensitive)
// ---------------------------------------------------------------------------
__global__ void __launch_bounds__(256)
qrot_scatter(const float* __restrict__ x,
             const float* __restrict__ axis,
             const float* __restrict__ theta,
             float* __restrict__ out) {
    __shared__ __align__(16) float tile[8192];

    const uint32_t tid  = threadIdx.x;
    const uint32_t wave = tid >> 5;       // wave32
    const uint32_t lane = tid & 31u;
    const uint32_t blockBase = blockIdx.x * 8192u;   // element index of block tile
    const uint32_t waveBase  = wave * 1024u;         // element offset of wave slice

    // ---- Tensor Data Mover: DMA this wave's 1024-float slice into LDS ------
    {
        const uint64_t gaddr = (uint64_t)(uintptr_t)(x + blockBase + waveBase);
        const uint32_t laddr = (uint32_t)(uintptr_t)(&tile[waveBase]); // LDS byte offset

        // D# group 0: count=1 | lds_addr | global_addr[56:0] | type=2
        const u32x4 g0 = {
            1u,                                                   // count=1, user mode
            laddr,                                                // lds_addr (bytes)
            (uint32_t)gaddr,                                      // global_addr[31:0]
            ((uint32_t)(gaddr >> 32) & 0x01FFFFFFu) | (2u << 30)  // addr[56:32] | type=2
        };
        // D# group 1: data_size=4B, 1-D tensor of 1024 elems, tile = 1024 elems
        const u32x8 g1 = {
            2u << 16,                    // workgroup_mask=0, data_size=2 (4 bytes)
            (1024u & 0xFFFFu) << 16,     // tensor_dim0[15:0]  -> bits 63:48
            (1024u >> 16) | (1u << 16),  // tensor_dim0[31:16] | tensor_dim1=1
            1024u << 16,                 // tile_dim0=1024     -> bits 127:112
            1u,                          // tile_dim1=1, tile_dim2=0
            1024u,                       // tensor_dim0_stride[31:0]
            0u,                          // stride hi / tensor_dim1_stride lo
            0u
        };
        asm volatile("tensor_load_to_lds %0, %1" :: "s"(g0), "s"(g1) : "memory");
#if __has_builtin(__builtin_amdgcn_s_wait_tensorcnt)
        __builtin_amdgcn_s_wait_tensorcnt(0);
#else
        asm volatile("s_wait_tensorcnt 0x0" ::: "memory");
#endif
    }

    // ---- Quaternion (uniform; precise transcendentals to match jnp) --------
    const float half = 0.5f * theta[0];
    const float sh   = sinf(half);
    const float qw   = cosf(half);
    const float qx   = axis[0] * sh;
    const float qy   = axis[1] * sh;
    const float qz   = axis[2] * sh;

    // ---- Process 8 float4 per lane, conflict-free LDS reads -----------------
    const float4* tile4 = reinterpret_cast<const float4*>(tile);
#pragma unroll
    for (int e = 0; e < 8; ++e) {
        const uint32_t f4   = wave * 256u + (uint32_t)e * 32u + lane; // float4 idx in tile
        const float4   v    = tile4[f4];
        const uint32_t base = blockBase + f4 * 4u;                    // global voxel idx
        const float*   vp   = &v.x;
#pragma unroll
        for (int c = 0; c < 4; ++c) {
#pragma clang fp contract(off)
            const float    occ = vp[c];
            const uint32_t g   = base + (uint32_t)c;
            const float fk = (float)(g & 255u);
            const float fj = (float)((g >> 8) & 255u);
            const float fi = (float)(g >> 16);

            // t = q * (0, i, j, k)   (Hamilton product, same term order as ref)
            const float tw = -(qx * fi) - qy * fj - qz * fk;
            const float tx =  qw * fi + qy * fk - qz * fj;
            const float ty =  qw * fj + qz * fi - qx * fk;
            const float tz =  qw * fk + qx * fj - qy * fi;
            // r = t * conj(q), vector part (identical FP sequence to reference)
            const float rx = -(tw * qx) + tx * qw - ty * qz + tz * qy;
            const float ry = -(tw * qy) + ty * qw - tz * qx + tx * qz;
            const float rz = -(tw * qz) + tz * qw - tx * qy + ty * qx;

            const int ix = (int)rx;   // trunc toward zero == .astype(int32)
            const int iy = (int)ry;
            const int iz = (int)rz;
            const bool inb = ((uint32_t)ix < 256u) & ((uint32_t)iy < 256u) &
                             ((uint32_t)iz < 256u);
            if (inb && occ != 0.0f) {
                // racy stores of the constant 1.0f are benign (matches scatter-set)
                out[((uint32_t)ix << 16) | ((uint32_t)iy << 8) | (uint32_t)iz] = 1.0f;
            }
        }
    }
}

// ---------------------------------------------------------------------------
extern "C" void kernel_launch(void* const* d_in, const int* in_sizes, int n_in,
                              void* d_out, int out_size, void* d_ws, size_t ws_size,
                              hipStream_t stream) {
    const float* x     = (const float*)d_in[0];   // [256,256,256] occupancy
    const float* axis  = (const float*)d_in[1];   // [3]
    const float* theta = (const float*)d_in[2];   // [1]
    float*       out   = (float*)d_out;           // [256,256,256]

    // 16,777,216 floats -> 4,194,304 float4 stores
    qrot_zero<<<16384, 256, 0, stream>>>(out);
    // 2048 blocks * 8192 voxels = 16,777,216 voxels
    qrot_scatter<<<2048, 256, 0, stream>>>(x, axis, theta, out);
}